// GCN_22411139350781
// MI455X (gfx1250) — compile-verified
//
#include <hip/hip_runtime.h>
#include <math.h>

typedef __attribute__((ext_vector_type(2))) float v2f;
typedef __attribute__((ext_vector_type(8))) float v8f;

// ---------------------------------------------------------------------------
// Small utility kernels
// ---------------------------------------------------------------------------
__global__ void gcn_fill_f32(float* __restrict__ p, float v, long long n) {
    long long i = (long long)blockIdx.x * blockDim.x + threadIdx.x;
    if (i < n) p[i] = v;
}

__global__ void gcn_deg_accum(const int* __restrict__ dst, float* __restrict__ deg, int E) {
    int e = blockIdx.x * blockDim.x + threadIdx.x;
    if (e < E) atomicAdd(&deg[dst[e]], 1.0f);
}

__global__ void gcn_rsqrt_inplace(float* __restrict__ d, int n) {
    int i = blockIdx.x * blockDim.x + threadIdx.x;
    if (i < n) d[i] = rsqrtf(d[i]);
}

// ---------------------------------------------------------------------------
// GEMM1: h1[N,64] = x[N,128] @ W1[128,64]   (full f32 via V_WMMA_F32_16X16X4_F32)
// block = 256 (8 waves). Each wave computes a 16-row x 64-col band (4 col tiles).
// A-frag (16x4 f32): row = lane&15; v0 holds K = 2*(lane>>4), v1 = that+1.
// B-frag (4x16): v0 rows K in {0,2}, v1 rows {1,3}, N = lane&15.
// C/D: VGPR r = rows r (lanes 0-15) / r+8 (lanes 16-31), N = lane&15.
// Out-of-range rows are CLAMPED (not masked) so loads stay unconditional and
// EXEC stays all-ones through the WMMA loop; stores remain guarded.
// ---------------------------------------------------------------------------
__global__ void gcn_gemm1_wmma(const float* __restrict__ x, const float* __restrict__ W1,
                               float* __restrict__ h1, int N) {
    const int lane = threadIdx.x & 31;
    const int wave = threadIdx.x >> 5;
    const int row0 = (blockIdx.x * 8 + wave) * 16;
    if (row0 >= N) return;                // uniform per wave
    const int m  = lane & 15;
    const int kh = (lane >> 4) * 2;       // 0 or 2 (half-wave K split)
    const int rowA = min(row0 + m, N - 1);
    const float2* __restrict__ xr = (const float2*)(x + (long long)rowA * 128);

    v8f acc0 = {}, acc1 = {}, acc2 = {}, acc3 = {};
    #pragma unroll 4
    for (int k = 0; k < 128; k += 4) {
        float2 a2 = xr[(k + kh) >> 1];    // 8-byte aligned: k%4==0, kh in {0,2}
        v2f a; a.x = a2.x; a.y = a2.y;

        const float* Wr0 = W1 + (long long)(k + kh) * 64 + m;
        const float* Wr1 = Wr0 + 64;
        v2f b0; b0.x = Wr0[0];  b0.y = Wr1[0];
        v2f b1; b1.x = Wr0[16]; b1.y = Wr1[16];
        v2f b2; b2.x = Wr0[32]; b2.y = Wr1[32];
        v2f b3; b3.x = Wr0[48]; b3.y = Wr1[48];

        acc0 = __builtin_amdgcn_wmma_f32_16x16x4_f32(false, a, false, b0, (short)0, acc0, false, false);
        acc1 = __builtin_amdgcn_wmma_f32_16x16x4_f32(false, a, false, b1, (short)0, acc1, false, false);
        acc2 = __builtin_amdgcn_wmma_f32_16x16x4_f32(false, a, false, b2, (short)0, acc2, false, false);
        acc3 = __builtin_amdgcn_wmma_f32_16x16x4_f32(false, a, false, b3, (short)0, acc3, false, false);
    }

    const int rbase = row0 + (lane >> 4) * 8;
    for (int r = 0; r < 8; ++r) {
        int rr = rbase + r;
        if (rr < N) {
            float* o = h1 + (long long)rr * 64 + m;
            o[0]  = acc0[r];
            o[16] = acc1[r];
            o[32] = acc2[r];
            o[48] = acc3[r];
        }
    }
}

// ---------------------------------------------------------------------------
// GEMM2: h2[N,40] = hrelu[N,64] @ W2[64,40]  (cols padded 40 -> 48, 3 tiles).
// Column indices >= 40 are clamped on load (garbage accumulates in cols 40-47
// but is never stored), keeping all loads unconditional.
// ---------------------------------------------------------------------------
__global__ void gcn_gemm2_wmma(const float* __restrict__ h, const float* __restrict__ W2,
                               float* __restrict__ h2, int N) {
    const int lane = threadIdx.x & 31;
    const int wave = threadIdx.x >> 5;
    const int row0 = (blockIdx.x * 8 + wave) * 16;
    if (row0 >= N) return;
    const int m  = lane & 15;
    const int kh = (lane >> 4) * 2;
    const int rowA = min(row0 + m, N - 1);
    const float2* __restrict__ hr = (const float2*)(h + (long long)rowA * 64);
    const int n2c = min(32 + m, 39);      // clamped col for tile 2

    v8f acc0 = {}, acc1 = {}, acc2 = {};
    #pragma unroll 4
    for (int k = 0; k < 64; k += 4) {
        float2 a2 = hr[(k + kh) >> 1];
        v2f a; a.x = a2.x; a.y = a2.y;

        const float* Wr0 = W2 + (long long)(k + kh) * 40;
        const float* Wr1 = Wr0 + 40;
        v2f b0; b0.x = Wr0[m];      b0.y = Wr1[m];
        v2f b1; b1.x = Wr0[16 + m]; b1.y = Wr1[16 + m];
        v2f b2; b2.x = Wr0[n2c];    b2.y = Wr1[n2c];

        acc0 = __builtin_amdgcn_wmma_f32_16x16x4_f32(false, a, false, b0, (short)0, acc0, false, false);
        acc1 = __builtin_amdgcn_wmma_f32_16x16x4_f32(false, a, false, b1, (short)0, acc1, false, false);
        acc2 = __builtin_amdgcn_wmma_f32_16x16x4_f32(false, a, false, b2, (short)0, acc2, false, false);
    }

    const int rbase = row0 + (lane >> 4) * 8;
    for (int r = 0; r < 8; ++r) {
        int rr = rbase + r;
        if (rr < N) {
            float* o = h2 + (long long)rr * 40;
            o[m]      = acc0[r];
            o[16 + m] = acc1[r];
            if (32 + m < 40) o[32 + m] = acc2[r];
        }
    }
}

// ---------------------------------------------------------------------------
// Edge scatter: agg[dst] += h[src] * dinv[src]*dinv[dst], float4 groups.
// F64 variant: feature dim 64 (16 float4 groups / edge).
// ---------------------------------------------------------------------------
__global__ void gcn_scatter64(const int* __restrict__ src, const int* __restrict__ dst,
                              const float* __restrict__ dinv, const float* __restrict__ h,
                              float* __restrict__ agg, long long total) {
    long long tid = (long long)blockIdx.x * blockDim.x + threadIdx.x;
    if (tid >= total) return;
    int e = (int)(tid >> 4);
    int f = (int)(tid & 15) * 4;
    int s = src[e], d = dst[e];
    float w = dinv[s] * dinv[d];
    const float4 v = *(const float4*)(h + (long long)s * 64 + f);
    float* o = agg + (long long)d * 64 + f;
    atomicAdd(o + 0, v.x * w);
    atomicAdd(o + 1, v.y * w);
    atomicAdd(o + 2, v.z * w);
    atomicAdd(o + 3, v.w * w);
}

// F40 variant: feature dim 40 (10 float4 groups / edge).
__global__ void gcn_scatter40(const int* __restrict__ src, const int* __restrict__ dst,
                              const float* __restrict__ dinv, const float* __restrict__ h,
                              float* __restrict__ agg, long long total) {
    long long tid = (long long)blockIdx.x * blockDim.x + threadIdx.x;
    if (tid >= total) return;
    int e = (int)(tid / 10);
    int f = (int)(tid % 10) * 4;
    int s = src[e], d = dst[e];
    float w = dinv[s] * dinv[d];
    const float4 v = *(const float4*)(h + (long long)s * 40 + f);
    float* o = agg + (long long)d * 40 + f;
    atomicAdd(o + 0, v.x * w);
    atomicAdd(o + 1, v.y * w);
    atomicAdd(o + 2, v.z * w);
    atomicAdd(o + 3, v.w * w);
}

// ---------------------------------------------------------------------------
// Self-loop + bias + ReLU, in place on agg1: agg1 = relu(agg1 + h1*dinv^2 + b1)
// ---------------------------------------------------------------------------
__global__ void gcn_self_bias_relu(float* __restrict__ agg1, const float* __restrict__ h1,
                                   const float* __restrict__ dinv, const float* __restrict__ b1,
                                   int N) {
    long long tid = (long long)blockIdx.x * blockDim.x + threadIdx.x;
    long long total = (long long)N * 64;
    if (tid >= total) return;
    int i = (int)(tid >> 6);
    int f = (int)(tid & 63);
    float di = dinv[i];
    float v = agg1[tid] + h1[tid] * (di * di) + b1[f];
    agg1[tid] = fmaxf(v, 0.0f);
}

// ---------------------------------------------------------------------------
// Finalize: z = out + h2*dinv^2 + b2, then row-wise log_softmax (one wave/row)
// out buffer already holds the edge-aggregated sums.
// ---------------------------------------------------------------------------
__global__ void gcn_finalize_logsoftmax(float* __restrict__ out, const float* __restrict__ h2,
                                        const float* __restrict__ dinv, const float* __restrict__ b2,
                                        int N) {
    const int lane = threadIdx.x & 31;
    const int wave = threadIdx.x >> 5;
    const int row  = blockIdx.x * 8 + wave;
    if (row >= N) return;
    float di = dinv[row];
    float di2 = di * di;
    const int f0 = lane, f1 = lane + 32;
    float z0 = -INFINITY, z1 = -INFINITY;
    if (f0 < 40) z0 = out[(long long)row * 40 + f0] + h2[(long long)row * 40 + f0] * di2 + b2[f0];
    if (f1 < 40) z1 = out[(long long)row * 40 + f1] + h2[(long long)row * 40 + f1] * di2 + b2[f1];

    float mx = fmaxf(z0, z1);
    #pragma unroll
    for (int off = 16; off > 0; off >>= 1) mx = fmaxf(mx, __shfl_xor(mx, off, 32));

    float s = 0.0f;
    if (f0 < 40) s += expf(z0 - mx);
    if (f1 < 40) s += expf(z1 - mx);
    #pragma unroll
    for (int off = 16; off > 0; off >>= 1) s += __shfl_xor(s, off, 32);

    float lse = mx + logf(s);
    if (f0 < 40) out[(long long)row * 40 + f0] = z0 - lse;
    if (f1 < 40) out[(long long)row * 40 + f1] = z1 - lse;
}

// ---------------------------------------------------------------------------
// Host-side launcher
// ---------------------------------------------------------------------------
static inline size_t align256(size_t x) { return (x + 255) & ~(size_t)255; }

extern "C" void kernel_launch(void* const* d_in, const int* in_sizes, int n_in,
                              void* d_out, int out_size, void* d_ws, size_t ws_size,
                              hipStream_t stream) {
    const float* x    = (const float*)d_in[0];
    const int*   ei   = (const int*)  d_in[1];
    const float* W1   = (const float*)d_in[2];
    const float* b1   = (const float*)d_in[3];
    const float* W2   = (const float*)d_in[4];
    const float* b2   = (const float*)d_in[5];
    float* out = (float*)d_out;

    const int N = in_sizes[0] / 128;   // 100000
    const int E = in_sizes[1] / 2;     // 1600000
    const int* src = ei;
    const int* dst = ei + E;

    // Workspace carving
    char* ws = (char*)d_ws;
    size_t off = 0;
    float* dinv = (float*)(ws + off); off += align256((size_t)N * sizeof(float));
    float* h1   = (float*)(ws + off); off += align256((size_t)N * 64 * sizeof(float));
    float* agg1 = (float*)(ws + off); off += align256((size_t)N * 64 * sizeof(float));
    float* h2   = (float*)(ws + off); off += align256((size_t)N * 40 * sizeof(float));
    (void)ws_size; (void)n_in; (void)out_size;

    const int B = 256;

    // 1) degree (with self loop) -> dinv = rsqrt(deg)
    gcn_fill_f32<<<(N + B - 1) / B, B, 0, stream>>>(dinv, 1.0f, N);
    gcn_deg_accum<<<(E + B - 1) / B, B, 0, stream>>>(dst, dinv, E);
    gcn_rsqrt_inplace<<<(N + B - 1) / B, B, 0, stream>>>(dinv, N);

    // 2) zero accumulation buffers (agg1 and d_out reused as agg2)
    long long n64 = (long long)N * 64;
    long long n40 = (long long)N * 40;
    gcn_fill_f32<<<(unsigned)((n64 + B - 1) / B), B, 0, stream>>>(agg1, 0.0f, n64);
    gcn_fill_f32<<<(unsigned)((n40 + B - 1) / B), B, 0, stream>>>(out, 0.0f, n40);

    // 3) h1 = x @ W1   (WMMA f32)
    int bands = (N + 16 * 8 - 1) / (16 * 8);
    gcn_gemm1_wmma<<<bands, B, 0, stream>>>(x, W1, h1, N);

    // 4) edge aggregation of h1 into agg1
    long long t1 = (long long)E * 16;
    gcn_scatter64<<<(unsigned)((t1 + B - 1) / B), B, 0, stream>>>(src, dst, dinv, h1, agg1, t1);

    // 5) self loop + bias + ReLU (in place -> hrelu in agg1)
    gcn_self_bias_relu<<<(unsigned)((n64 + B - 1) / B), B, 0, stream>>>(agg1, h1, dinv, b1, N);

    // 6) h2 = hrelu @ W2  (WMMA f32, 40 cols padded to 48)
    gcn_gemm2_wmma<<<bands, B, 0, stream>>>(agg1, W2, h2, N);

    // 7) edge aggregation of h2 into d_out
    long long t2 = (long long)E * 10;
    gcn_scatter40<<<(unsigned)((t2 + B - 1) / B), B, 0, stream>>>(src, dst, dinv, h2, out, t2);

    // 8) self loop + bias + log_softmax (one wave per row, in place on d_out)
    gcn_finalize_logsoftmax<<<(N + 7) / 8, B, 0, stream>>>(out, h2, dinv, b2, N);
}